// GCNClassifier_90305982366169
// MI455X (gfx1250) — compile-verified
//
#include <hip/hip_runtime.h>

// ---------------------------------------------------------------------------
// GCN forward for MI455X (gfx1250), FP32 WMMA (V_WMMA_F32_16X16X4_F32).
// Compute-bound in FP32 (~52 GFLOP vs ~6 us of HBM traffic at 23.3 TB/s).
// Wave tile 32x64 (2x4 frag tiles): 6 LDS frag loads feed 8 WMMAs per K-step.
// ---------------------------------------------------------------------------

#define HEADS 3
#define LAYERS 2
#define MEMD  256
#define NB    32          // batch
#define SS    512         // sequence
#define FDIM  (MEMD * HEADS * LAYERS)   // 1536

typedef __attribute__((ext_vector_type(2))) float v2f;
typedef __attribute__((ext_vector_type(4))) float v4f;
typedef __attribute__((ext_vector_type(8))) float v8f;

__device__ __forceinline__ v8f wmma_f32(v2f a, v2f b, v8f c) {
    // D = A(16x4,f32) * B(4x16,f32) + C(16x16,f32)
    return __builtin_amdgcn_wmma_f32_16x16x4_f32(
        false, a, false, b, (short)0, c, false, false);
}

// ---------------------------------------------------------------------------
// Kernel 1: recip[h,b,s] = 1 / (sum_t adj[h,b,s,t] + 1).  One wave per row.
// ---------------------------------------------------------------------------
__global__ void denom_recip_kernel(const float* __restrict__ adj,
                                   float* __restrict__ recip) {
    int lane = threadIdx.x & 31;
    int wv   = threadIdx.x >> 5;
    size_t row = (size_t)blockIdx.x * 8 + wv;      // HEADS*NB*SS rows total
    const v4f* p = (const v4f*)(adj + row * SS);
    float s = 0.f;
    #pragma unroll
    for (int i = 0; i < SS / 128; ++i) {           // 4 x float4 per lane
        v4f v = p[lane + i * 32];
        s += v[0] + v[1] + v[2] + v[3];
    }
    #pragma unroll
    for (int off = 16; off > 0; off >>= 1) s += __shfl_xor(s, off, 32);
    if (lane == 0) recip[row] = 1.0f / (s + 1.0f);
}

// ---------------------------------------------------------------------------
// Kernel 2: fused GCN layer for one (head, layer).
//   Y = relu( ((adj @ X + X) @ W^T + 2*b) * recip )   scattered into `final`.
// Block: 256 threads = 8 waves.  Block tile: 64 rows x 256 cols.
// Wave w: rows 32*(w>>2) (2 row-tiles), cols 64*(w&3) (4 col-tiles).
// Grid: (SS/64, NB).
// ---------------------------------------------------------------------------
#define KC   32
#define LDA  36     // 64  x (KC+4)  A staging, rows 16B aligned, banks distinct
#define LDB  36     // 256 x (KC+4)  B staging (X^T tile, then W tile)
#define LDS2 258    // 64  x 258     (Ax + X) parked for phase 2

__global__ void gcn_layer_kernel(const float* __restrict__ adj_h,   // [NB,SS,SS]
                                 const float* __restrict__ Xin,
                                 long long xRowStride,
                                 long long xBatchStride,
                                 const float* __restrict__ Wl,      // [256,256] [out][in]
                                 const float* __restrict__ bl,      // [256]
                                 const float* __restrict__ rden,    // [NB,SS] 1/denom
                                 float* __restrict__ Yout,          // final + colOff
                                 long long yRowStride,
                                 long long yBatchStride) {
    extern __shared__ float smem[];
    float* As1 = smem;                 // 64*LDA
    float* Bs  = As1 + 64 * LDA;       // 256*LDB
    float* S2  = Bs  + 256 * LDB;      // 64*LDS2

    const int tid  = threadIdx.x;
    const int lane = tid & 31;
    const int wv   = tid >> 5;
    const int bidx = blockIdx.y;
    const int r0   = blockIdx.x * 64;

    const int rw = wv >> 2;            // 0..1 : rows 32*rw
    const int cw = wv & 3;             // 0..3 : cols 64*cw
    const int am = lane & 15;          // m (A) / n (B) within a 16-tile
    const int kh = lane >> 4;          // lane half
    const int ak = kh * 2;             // k sub-offset within fragment

    const float* A = adj_h + (size_t)bidx * SS * SS;
    const float* X = Xin   + (size_t)bidx * xBatchStride;

    // wave-constant LDS fragment base pointers (all tile offsets are
    // compile-time constants -> fold into ds_load instruction offsets)
    const float* pA1 = &As1[(rw * 32 + am) * LDA + ak];
    const float* pB  = &Bs [(cw * 64 + am) * LDB + ak];
    const float* pA2 = &S2 [(rw * 32 + am) * LDS2 + ak];

    const v8f vzero = {};
    v8f acc[8];                        // [rt*4 + ct]
    #pragma unroll
    for (int i = 0; i < 8; ++i) acc[i] = vzero;

    // -------- Phase 1: Ax tile = adj[r0:r0+64, :] @ X[:, 0:256] ------------
    for (int k0 = 0; k0 < SS; k0 += KC) {
        #pragma unroll
        for (int i = 0; i < 2; ++i) {               // adj tile 64x32, float4
            int q = i * 256 + tid;
            int rr = q >> 3, kq = (q & 7) * 4;
            *(v4f*)&As1[rr * LDA + kq] =
                *(const v4f*)&A[(size_t)(r0 + rr) * SS + k0 + kq];
        }
        #pragma unroll
        for (int i = 0; i < 8; ++i) {               // X^T tile via reg transpose
            int kk = i * 4 + (tid >> 6);            // 0..31
            int n0 = (tid & 63) * 4;                // 0..252
            v4f v = *(const v4f*)&X[(long long)(k0 + kk) * xRowStride + n0];
            Bs[(n0 + 0) * LDB + kk] = v[0];
            Bs[(n0 + 1) * LDB + kk] = v[1];
            Bs[(n0 + 2) * LDB + kk] = v[2];
            Bs[(n0 + 3) * LDB + kk] = v[3];
        }
        __syncthreads();
        #pragma unroll
        for (int kk = 0; kk < KC; kk += 4) {
            v2f a0 = *(const v2f*)(pA1 + kk);
            v2f a1 = *(const v2f*)(pA1 + 16 * LDA + kk);
            v2f b0 = *(const v2f*)(pB + kk);
            v2f b1 = *(const v2f*)(pB + 16 * LDB + kk);
            v2f b2 = *(const v2f*)(pB + 32 * LDB + kk);
            v2f b3 = *(const v2f*)(pB + 48 * LDB + kk);
            acc[0] = wmma_f32(a0, b0, acc[0]);
            acc[1] = wmma_f32(a0, b1, acc[1]);
            acc[2] = wmma_f32(a0, b2, acc[2]);
            acc[3] = wmma_f32(a0, b3, acc[3]);
            acc[4] = wmma_f32(a1, b0, acc[4]);
            acc[5] = wmma_f32(a1, b1, acc[5]);
            acc[6] = wmma_f32(a1, b2, acc[6]);
            acc[7] = wmma_f32(a1, b3, acc[7]);
        }
        __syncthreads();
    }

    // -------- park (Ax + X) in LDS as A-operand for phase 2 ----------------
    #pragma unroll
    for (int rt = 0; rt < 2; ++rt) {
        #pragma unroll
        for (int ct = 0; ct < 4; ++ct) {
            int n = cw * 64 + ct * 16 + am;
            #pragma unroll
            for (int r = 0; r < 8; ++r) {
                int row = rw * 32 + rt * 16 + r + 8 * kh;
                float x = X[(size_t)(r0 + row) * xRowStride + n];
                S2[row * LDS2 + n] = acc[rt * 4 + ct][r] + x;
            }
        }
    }
    __syncthreads();

    // -------- Phase 2: Y = (Ax+X) @ W^T ------------------------------------
    #pragma unroll
    for (int i = 0; i < 8; ++i) acc[i] = vzero;

    for (int k0 = 0; k0 < MEMD; k0 += KC) {
        #pragma unroll
        for (int i = 0; i < 8; ++i) {               // W tile, float4 both sides
            int q = i * 256 + tid;
            int n = q >> 3, kq = (q & 7) * 4;
            *(v4f*)&Bs[n * LDB + kq] =
                *(const v4f*)&Wl[(size_t)n * MEMD + k0 + kq];
        }
        __syncthreads();
        #pragma unroll
        for (int kk = 0; kk < KC; kk += 4) {
            v2f a0 = *(const v2f*)(pA2 + k0 + kk);
            v2f a1 = *(const v2f*)(pA2 + 16 * LDS2 + k0 + kk);
            v2f b0 = *(const v2f*)(pB + kk);
            v2f b1 = *(const v2f*)(pB + 16 * LDB + kk);
            v2f b2 = *(const v2f*)(pB + 32 * LDB + kk);
            v2f b3 = *(const v2f*)(pB + 48 * LDB + kk);
            acc[0] = wmma_f32(a0, b0, acc[0]);
            acc[1] = wmma_f32(a0, b1, acc[1]);
            acc[2] = wmma_f32(a0, b2, acc[2]);
            acc[3] = wmma_f32(a0, b3, acc[3]);
            acc[4] = wmma_f32(a1, b0, acc[4]);
            acc[5] = wmma_f32(a1, b1, acc[5]);
            acc[6] = wmma_f32(a1, b2, acc[6]);
            acc[7] = wmma_f32(a1, b3, acc[7]);
        }
        __syncthreads();
    }

    // -------- epilogue: +2b, /denom, relu, scatter into `final` ------------
    const float* rd = rden + (size_t)bidx * SS + r0;
    float* Y = Yout + (size_t)bidx * yBatchStride;
    #pragma unroll
    for (int rt = 0; rt < 2; ++rt) {
        #pragma unroll
        for (int ct = 0; ct < 4; ++ct) {
            int n = cw * 64 + ct * 16 + am;
            float bias2 = 2.0f * bl[n];
            #pragma unroll
            for (int r = 0; r < 8; ++r) {
                int row = rw * 32 + rt * 16 + r + 8 * kh;
                float v = (acc[rt * 4 + ct][r] + bias2) * rd[row];
                Y[(size_t)(r0 + row) * yRowStride + n] = v > 0.f ? v : 0.f;
            }
        }
    }
}

// ---------------------------------------------------------------------------
// Kernel 3: out = gcn_inputs + final @ W_out^T + b_out.  M=NB*SS,K=1536,N=256.
// Same 64x256 block tile / 2x4 wave tile.  Grid: NB*SS/64.
// ---------------------------------------------------------------------------
__global__ void out_linear_kernel(const float* __restrict__ Fin,   // [NB*SS, FDIM]
                                  const float* __restrict__ Wout,  // [256, FDIM]
                                  const float* __restrict__ bout,  // [256]
                                  const float* __restrict__ Xg,    // [NB*SS, 256]
                                  float* __restrict__ out) {       // [NB*SS, 256]
    extern __shared__ float smem[];
    float* As = smem;                 // 64*LDA
    float* Bs = As + 64 * LDA;        // 256*LDB

    const int tid  = threadIdx.x;
    const int lane = tid & 31;
    const int wv   = tid >> 5;
    const int r0   = blockIdx.x * 64;

    const int rw = wv >> 2;
    const int cw = wv & 3;
    const int am = lane & 15;
    const int kh = lane >> 4;
    const int ak = kh * 2;

    const float* pA = &As[(rw * 32 + am) * LDA + ak];
    const float* pB = &Bs[(cw * 64 + am) * LDB + ak];

    const v8f vzero = {};
    v8f acc[8];
    #pragma unroll
    for (int i = 0; i < 8; ++i) acc[i] = vzero;

    for (int k0 = 0; k0 < FDIM; k0 += KC) {
        #pragma unroll
        for (int i = 0; i < 2; ++i) {               // A tile 64x32, float4
            int q = i * 256 + tid;
            int rr = q >> 3, kq = (q & 7) * 4;
            *(v4f*)&As[rr * LDA + kq] =
                *(const v4f*)&Fin[(size_t)(r0 + rr) * FDIM + k0 + kq];
        }
        #pragma unroll
        for (int i = 0; i < 8; ++i) {               // W_out tile, float4
            int q = i * 256 + tid;
            int n = q >> 3, kq = (q & 7) * 4;
            *(v4f*)&Bs[n * LDB + kq] =
                *(const v4f*)&Wout[(size_t)n * FDIM + k0 + kq];
        }
        __syncthreads();
        #pragma unroll
        for (int kk = 0; kk < KC; kk += 4) {
            v2f a0 = *(const v2f*)(pA + kk);
            v2f a1 = *(const v2f*)(pA + 16 * LDA + kk);
            v2f b0 = *(const v2f*)(pB + kk);
            v2f b1 = *(const v2f*)(pB + 16 * LDB + kk);
            v2f b2 = *(const v2f*)(pB + 32 * LDB + kk);
            v2f b3 = *(const v2f*)(pB + 48 * LDB + kk);
            acc[0] = wmma_f32(a0, b0, acc[0]);
            acc[1] = wmma_f32(a0, b1, acc[1]);
            acc[2] = wmma_f32(a0, b2, acc[2]);
            acc[3] = wmma_f32(a0, b3, acc[3]);
            acc[4] = wmma_f32(a1, b0, acc[4]);
            acc[5] = wmma_f32(a1, b1, acc[5]);
            acc[6] = wmma_f32(a1, b2, acc[6]);
            acc[7] = wmma_f32(a1, b3, acc[7]);
        }
        __syncthreads();
    }

    #pragma unroll
    for (int rt = 0; rt < 2; ++rt) {
        #pragma unroll
        for (int ct = 0; ct < 4; ++ct) {
            int n = cw * 64 + ct * 16 + am;
            float bb = bout[n];
            #pragma unroll
            for (int r = 0; r < 8; ++r) {
                int row = r0 + rw * 32 + rt * 16 + r + 8 * kh;
                out[(size_t)row * MEMD + n] =
                    acc[rt * 4 + ct][r] + bb + Xg[(size_t)row * MEMD + n];
            }
        }
    }
}

// ---------------------------------------------------------------------------
extern "C" void kernel_launch(void* const* d_in, const int* in_sizes, int n_in,
                              void* d_out, int out_size, void* d_ws, size_t ws_size,
                              hipStream_t stream) {
    const float* adj  = (const float*)d_in[0];   // [3,32,512,512]
    const float* gcn  = (const float*)d_in[1];   // [32,512,256]
    // d_in[2] mask, d_in[3] domain_mask: unused by the reference forward
    const float* W    = (const float*)d_in[4];   // [6,256,256]
    const float* bvec = (const float*)d_in[5];   // [6,256]
    const float* Wout = (const float*)d_in[6];   // [256,1536]
    const float* bout = (const float*)d_in[7];   // [256]
    float* out = (float*)d_out;

    // workspace: recip [3*32*512] | final [32*512*1536]
    float* recip    = (float*)d_ws;
    float* finalbuf = recip + (size_t)HEADS * NB * SS;

    // 1) per-head 1/denom
    denom_recip_kernel<<<(HEADS * NB * SS) / 8, 256, 0, stream>>>(adj, recip);

    // 2) six fused GCN layers -> scatter into `final` (concat layout)
    const size_t ldsFused = (size_t)(64 * LDA + 256 * LDB + 64 * LDS2) * sizeof(float);
    for (int h = 0; h < HEADS; ++h) {
        for (int l = 0; l < LAYERS; ++l) {
            int idx = h * LAYERS + l;
            const float* adj_h = adj + (size_t)h * NB * SS * SS;
            const float* Xin;
            long long xrs, xbs;
            if (l == 0) { Xin = gcn; xrs = MEMD; xbs = (long long)SS * MEMD; }
            else        { Xin = finalbuf + (size_t)(idx - 1) * MEMD;
                          xrs = FDIM; xbs = (long long)SS * FDIM; }
            float* Y = finalbuf + (size_t)idx * MEMD;
            dim3 grid(SS / 64, NB);
            gcn_layer_kernel<<<grid, 256, ldsFused, stream>>>(
                adj_h, Xin, xrs, xbs,
                W + (size_t)idx * MEMD * MEMD, bvec + (size_t)idx * MEMD,
                recip + (size_t)h * NB * SS,
                Y, (long long)FDIM, (long long)SS * FDIM);
        }
    }

    // 3) output linear + residual
    const size_t ldsOut = (size_t)(64 * LDA + 256 * LDB) * sizeof(float);
    out_linear_kernel<<<(NB * SS) / 64, 256, ldsOut, stream>>>(
        finalbuf, Wout, bout, gcn, out);
}